// CTM_44976897523878
// MI455X (gfx1250) — compile-verified
//
#include <hip/hip_runtime.h>

typedef __attribute__((ext_vector_type(2))) float v2f;
typedef __attribute__((ext_vector_type(8))) float v8f;

#define S_DIM    2048
#define M_DIM    64
#define B_DIM    16
#define CH_DIM   682
#define NOUT_DIM 128
#define T_DIM    16

// ---------------------------------------------------------------------------
// Init: post_act[b][s] = i_post_act[s]; ring[m][b][s] = i_pre_act_mem[s][m]
// ---------------------------------------------------------------------------
__global__ void CTM_init_state(const float* __restrict__ i_post_act,
                               const float* __restrict__ i_pre_act_mem,
                               float* __restrict__ post_act,
                               float* __restrict__ ring) {
    int tid    = blockIdx.x * blockDim.x + threadIdx.x;
    int stride = gridDim.x * blockDim.x;
    const int RING_N = M_DIM * B_DIM * S_DIM;
    for (int idx = tid; idx < RING_N; idx += stride) {
        int s  = idx % S_DIM;
        int bm = idx / S_DIM;
        int m  = bm / B_DIM;
        ring[idx] = i_pre_act_mem[s * M_DIM + m];
    }
    for (int idx = tid; idx < B_DIM * S_DIM; idx += stride) {
        int s = idx % S_DIM;
        post_act[idx] = i_post_act[s];
    }
}

// ---------------------------------------------------------------------------
// pre_act(16 x 2048) = post_act(16 x 2048) @ W_syn^T + b_syn
// One 16x16 output tile per block; K=2048 split over 4 waves, each wave runs
// V_WMMA_F32_16X16X4_F32 over its K range, partials reduced through LDS.
// A[m,k] = post_act[m][k]   (lanes 0-15 -> M, VGPR pair -> K + 2*laneHalf)
// B[k,n] = W_syn[s0+n][k]   (lanes -> N columns, VGPR pair -> K + 2*laneHalf)
// K-offset of the wave is folded into the pointers so the inner loop has a
// compile-time trip count -> no EXEC-masked loop around the WMMAs.
// ---------------------------------------------------------------------------
__global__ __launch_bounds__(128) void CTM_syn_gemm(
        const float* __restrict__ post_act,
        const float* __restrict__ W_syn,
        const float* __restrict__ b_syn,
        float* __restrict__ pre_act) {
    __shared__ float red[4 * 256];

    const int lane = threadIdx.x & 31;
    const int wave = threadIdx.x >> 5;
    const int s0   = blockIdx.x * 16;
    const int n    = lane & 15;          // A: batch row / B: S column
    const int koff = (lane >> 4) << 1;   // 0 or 2
    const int kbeg = wave * (S_DIM / 4); // 512-wide K slice per wave

    const float* pa = post_act + (size_t)n * S_DIM + kbeg + koff;
    const float* pb = W_syn + (size_t)(s0 + n) * S_DIM + kbeg + koff;

    v8f c = {0.f, 0.f, 0.f, 0.f, 0.f, 0.f, 0.f, 0.f};

#pragma unroll 4
    for (int kk = 0; kk < S_DIM / 4; kk += 4) {
        // prefetch the W_syn stream one 512B chunk ahead (global_prefetch_b8)
        if ((kk & 31) == 0) __builtin_prefetch(pb + kk + 128, 0, 1);
        v2f a = *(const v2f*)(pa + kk);
        v2f b = *(const v2f*)(pb + kk);
        c = __builtin_amdgcn_wmma_f32_16x16x4_f32(
                /*neg_a=*/false, a, /*neg_b=*/false, b,
                /*c_mod=*/(short)0, c, /*reuse_a=*/false, /*reuse_b=*/false);
    }

#pragma unroll
    for (int v = 0; v < 8; ++v) red[wave * 256 + v * 32 + lane] = c[v];
    __syncthreads();

    // 256 tile elements, 128 threads -> 2 each; e -> (row, col) per D layout
    for (int e = threadIdx.x; e < 256; e += 128) {
        float sum = red[e] + red[256 + e] + red[512 + e] + red[768 + e];
        int   l2  = e & 31;
        int   row = (e >> 5) + 8 * ((l2 >> 4) & 1);   // batch index
        int   col = l2 & 15;                          // S offset in tile
        pre_act[(size_t)row * S_DIM + s0 + col] = sum + b_syn[s0 + col];
    }
}

// ---------------------------------------------------------------------------
// Ring append + post_act[b,s] = sum_m ring[(t+1+m)&63][b][s] * W_nlm[s][m] + b_nlm[s]
// (m=63 term is the freshly computed pre_act, taken from register.)
// ---------------------------------------------------------------------------
__global__ void CTM_nlm_step(const float* __restrict__ pre_act,
                             float* __restrict__ ring,
                             const float* __restrict__ W_nlm,
                             const float* __restrict__ b_nlm,
                             float* __restrict__ post_act,
                             int t) {
    int tid = blockIdx.x * blockDim.x + threadIdx.x;   // over B*S
    if (tid >= B_DIM * S_DIM) return;
    int s = tid % S_DIM;
    int b = tid / S_DIM;

    const float  nv = pre_act[tid];
    const float* wn = W_nlm + (size_t)s * M_DIM;
    const int    base  = t + 1;
    const size_t plane = (size_t)B_DIM * S_DIM;
    const size_t bs    = (size_t)b * S_DIM + s;

    float acc = b_nlm[s];
#pragma unroll 7
    for (int m = 0; m < 63; ++m) {
        int phys = (base + m) & 63;
        acc += ring[phys * plane + bs] * wn[m];
    }
    acc += nv * wn[63];

    ring[(size_t)(t & 63) * plane + bs] = nv;
    post_act[tid] = acc;
}

// ---------------------------------------------------------------------------
// l[b,i] = post_act[b, idx_left[i]]
// rw[b,o] = sum_j post_act[b, idx_right[j]] * W_out[o,j]
// ---------------------------------------------------------------------------
__global__ void CTM_gather_rw(const float* __restrict__ post_act,
                              const int* __restrict__ idx_left,
                              const int* __restrict__ idx_right,
                              const float* __restrict__ W_out,
                              float* __restrict__ lbuf,
                              float* __restrict__ rwbuf) {
    int tid = blockIdx.x * blockDim.x + threadIdx.x;
    if (tid < B_DIM * CH_DIM) {
        int i = tid % CH_DIM;
        int b = tid / CH_DIM;
        lbuf[tid] = post_act[(size_t)b * S_DIM + idx_left[i]];
    }
    int t2 = tid - B_DIM * CH_DIM;
    if (t2 >= 0 && t2 < B_DIM * NOUT_DIM) {
        int o = t2 % NOUT_DIM;
        int b = t2 / NOUT_DIM;
        const float* wo = W_out + (size_t)o * CH_DIM;
        const float* pb = post_act + (size_t)b * S_DIM;
        float acc = 0.f;
        for (int j = 0; j < CH_DIM; ++j) acc += pb[idx_right[j]] * wo[j];
        rwbuf[t2] = acc;
    }
}

// ---------------------------------------------------------------------------
// out[t] = (t==0 ? b_out : out[t-1]) + d2 * l ⊗ rw   (float4 over NOUT)
// ---------------------------------------------------------------------------
__global__ void CTM_rank1_out(const float* __restrict__ lbuf,
                              const float* __restrict__ rwbuf,
                              const float* __restrict__ b_out,
                              const float* __restrict__ decay,
                              float* __restrict__ out,
                              int t) {
    const int O4  = NOUT_DIM / 4;
    const int TOT = B_DIM * CH_DIM * O4;
    int tid = blockIdx.x * blockDim.x + threadIdx.x;
    if (tid >= TOT) return;

    float d2 = decay[0] * decay[0];
    int o4 = (tid % O4) * 4;
    int bi = tid / O4;
    int i  = bi % CH_DIM;
    int b  = bi / CH_DIM;

    float lv = lbuf[b * CH_DIM + i] * d2;
    float4 rw = *(const float4*)(rwbuf + b * NOUT_DIM + o4);

    size_t cur = ((((size_t)t * B_DIM + b) * CH_DIM + i) * NOUT_DIM) + o4;
    float4 prev;
    if (t == 0) prev = *(const float4*)(b_out + o4);
    else        prev = *(const float4*)(out + cur - (size_t)B_DIM * CH_DIM * NOUT_DIM);

    float4 res;
    res.x = prev.x + lv * rw.x;
    res.y = prev.y + lv * rw.y;
    res.z = prev.z + lv * rw.z;
    res.w = prev.w + lv * rw.w;
    *(float4*)(out + cur) = res;
}

// ---------------------------------------------------------------------------
extern "C" void kernel_launch(void* const* d_in, const int* in_sizes, int n_in,
                              void* d_out, int out_size, void* d_ws, size_t ws_size,
                              hipStream_t stream) {
    const float* W_syn      = (const float*)d_in[1];
    const float* b_syn      = (const float*)d_in[2];
    const float* W_nlm      = (const float*)d_in[3];
    const float* b_nlm      = (const float*)d_in[4];
    const float* decay      = (const float*)d_in[5];
    const float* W_out      = (const float*)d_in[6];
    const float* b_out      = (const float*)d_in[7];
    const float* i_post_act = (const float*)d_in[8];
    const float* i_pre_mem  = (const float*)d_in[9];
    const int*   idx_left   = (const int*)d_in[10];
    const int*   idx_right  = (const int*)d_in[11];

    float* ws       = (float*)d_ws;
    float* post_act = ws;                        // 32768 floats
    float* pre_act  = ws + 32768;                // 32768 floats
    float* lbuf     = ws + 65536;                // 10912 floats
    float* rwbuf    = ws + 76448;                // 2048 floats (16B aligned)
    float* ring     = ws + 81920;                // 64*16*2048 = 2,097,152 floats

    float* out = (float*)d_out;

    CTM_init_state<<<512, 256, 0, stream>>>(i_post_act, i_pre_mem, post_act, ring);

    const int NLM_BLOCKS = (B_DIM * S_DIM) / 256;                       // 128
    const int GR_BLOCKS  = (B_DIM * CH_DIM + B_DIM * NOUT_DIM + 255) / 256;
    const int R1_BLOCKS  = (B_DIM * CH_DIM * (NOUT_DIM / 4) + 255) / 256;

    for (int t = 0; t < T_DIM; ++t) {
        CTM_syn_gemm<<<S_DIM / 16, 128, 0, stream>>>(post_act, W_syn, b_syn, pre_act);
        CTM_nlm_step<<<NLM_BLOCKS, 256, 0, stream>>>(pre_act, ring, W_nlm, b_nlm,
                                                     post_act, t);
        CTM_gather_rw<<<GR_BLOCKS, 256, 0, stream>>>(post_act, idx_left, idx_right,
                                                     W_out, lbuf, rwbuf);
        CTM_rank1_out<<<R1_BLOCKS, 256, 0, stream>>>(lbuf, rwbuf, b_out, decay,
                                                     out, t);
    }
}